// GraphSAGE_16200616640670
// MI455X (gfx1250) — compile-verified
//
#include <hip/hip_runtime.h>

#define N_NODES 100000
#define FDIM 128
#define NCLS 7
#define ASTRIDE 260   // 16-row LDS pitch (floats); 260 % 64 == 4 -> conflict-free A reads

static_assert(N_NODES % 16 == 0, "no tail tile: row guards removed in k_sage");
static_assert((ASTRIDE * 4) % 16 == 0, "16B-aligned LDS rows for b128 stores");

typedef float v2f __attribute__((ext_vector_type(2)));
typedef float v8f __attribute__((ext_vector_type(8)));

// ---------------- zero workspace ----------------
__global__ void k_zero_f4(float4* p, long n4) {
    long i = blockIdx.x * (long)blockDim.x + threadIdx.x;
    if (i < n4) p[i] = make_float4(0.f, 0.f, 0.f, 0.f);
}

// ---------------- degree count ----------------
__global__ void k_deg(const int* __restrict__ dst, float* __restrict__ deg, int E) {
    int e = blockIdx.x * blockDim.x + threadIdx.x;
    if (e < E) atomicAdd(&deg[dst[e]], 1.0f);
}

// ---------------- edge scatter: agg[dst] += feat[src] ----------------
// one thread per (edge, 4-float chunk): 32 threads cover one 128-float row
__global__ void k_scatter(const float* __restrict__ feat, const int* __restrict__ src,
                          const int* __restrict__ dst, float* __restrict__ agg, int E) {
    long gid = blockIdx.x * (long)blockDim.x + threadIdx.x;
    int e = (int)(gid >> 5);
    if (e >= E) return;
    int c = ((int)gid & 31) * 4;
    int s = src[e];
    int d = dst[e];
    const float4 v = *(const float4*)(feat + (long)s * FDIM + c);
    float* o = agg + (long)d * FDIM + c;
    atomicAdd(o + 0, v.x);
    atomicAdd(o + 1, v.y);
    atomicAdd(o + 2, v.z);
    atomicAdd(o + 3, v.w);
}

// ---------------- fused SAGE layer: out = relu([agg/deg | xf] @ [Wl;Wr]^T + b) ----------------
// block = 256 threads (8 waves); block handles 16 node rows; wave w handles cols [16w,16w+16)
__global__ __launch_bounds__(256) void k_sage(
    const float* __restrict__ agg, const float* __restrict__ deg,
    const float* __restrict__ xf,
    const float* __restrict__ Wl, const float* __restrict__ Wr,
    const float* __restrict__ bias, float* __restrict__ out) {
    __shared__ float A[16 * ASTRIDE];
    const int t = threadIdx.x;
    const long row0 = (long)blockIdx.x * 16;

    // stage A = [agg/deg | xf], 16 rows x 256 K-values, float4 loads + b128 LDS stores
#pragma unroll
    for (int i = 0; i < 4; ++i) {
        const int idx = i * 256 + t;        // 0..1023 -> (row, 4-float chunk)
        const int row = idx >> 6;           // 16 rows
        const int c4  = (idx & 63) * 4;     // 0..252
        const long node = row0 + row;
        float4 v;
        if (c4 < FDIM) {
            const float invd = 1.0f / fmaxf(deg[node], 1.0f);
            v = *(const float4*)(agg + node * FDIM + c4);
            v.x *= invd; v.y *= invd; v.z *= invd; v.w *= invd;
        } else {
            v = *(const float4*)(xf + node * FDIM + (c4 - FDIM));
        }
        *(float4*)&A[row * ASTRIDE + c4] = v;
    }
    __syncthreads();

    const int lane = t & 31;
    const int wave = t >> 5;
    const int half = lane >> 4;   // 0: lanes 0-15, 1: lanes 16-31
    const int lrow = lane & 15;
    const int col  = wave * 16 + lrow;

    v8f acc = {0.f, 0.f, 0.f, 0.f, 0.f, 0.f, 0.f, 0.f};
    for (int k0 = 0; k0 < 2 * FDIM; k0 += 4) {
        const int ka = k0 + 2 * half;      // A: lanes 0-15 hold K=k0,k0+1; lanes 16-31 K=k0+2,k0+3
        v2f a;
        a.x = A[lrow * ASTRIDE + ka];
        a.y = A[lrow * ASTRIDE + ka + 1];
        // B[k][col] = W[col][k]; mirrored layout (lanes 0-15: rows k0,k0+1)
        const float* Wm = (k0 < FDIM) ? Wl : Wr;
        const int km = ka & (FDIM - 1);
        v2f b;
        b.x = Wm[col * FDIM + km];
        b.y = Wm[col * FDIM + km + 1];
        acc = __builtin_amdgcn_wmma_f32_16x16x4_f32(
            /*neg_a=*/false, a, /*neg_b=*/false, b,
            /*c_mod=*/(short)0, acc, /*reuse_a=*/false, /*reuse_b=*/false);
    }

    // C/D layout: VGPR r -> M = r (+8 for lanes 16-31); N_NODES % 16 == 0 -> no guards
    const float bb = bias[col];
    float* obase = out + (row0 + 8 * half) * FDIM + col;
#pragma unroll
    for (int r = 0; r < 8; ++r) {
        obase[r * FDIM] = fmaxf(acc[r] + bb, 0.f);
    }
}

// ---------------- classifier + log_softmax: one wave per node ----------------
__global__ __launch_bounds__(256) void k_cls(const float* __restrict__ h,
                                             const float* __restrict__ W,
                                             const float* __restrict__ b,
                                             float* __restrict__ out) {
    long gid = blockIdx.x * (long)blockDim.x + threadIdx.x;
    int node = (int)(gid >> 5);
    int lane = (int)gid & 31;
    if (node >= N_NODES) return;
    const float4 v = *(const float4*)(h + (long)node * FDIM + lane * 4);
    float lg[NCLS];
#pragma unroll
    for (int c = 0; c < NCLS; ++c) {
        const float4 w = *(const float4*)(W + c * FDIM + lane * 4);
        float p = v.x * w.x + v.y * w.y + v.z * w.z + v.w * w.w;
#pragma unroll
        for (int m = 16; m >= 1; m >>= 1) p += __shfl_xor(p, m, 32);
        lg[c] = p + b[c];
    }
    if (lane == 0) {
        float mx = lg[0];
#pragma unroll
        for (int c = 1; c < NCLS; ++c) mx = fmaxf(mx, lg[c]);
        float s = 0.f;
#pragma unroll
        for (int c = 0; c < NCLS; ++c) s += __expf(lg[c] - mx);
        float lse = mx + __logf(s);
#pragma unroll
        for (int c = 0; c < NCLS; ++c) out[(long)node * NCLS + c] = lg[c] - lse;
    }
}

extern "C" void kernel_launch(void* const* d_in, const int* in_sizes, int n_in,
                              void* d_out, int out_size, void* d_ws, size_t ws_size,
                              hipStream_t stream) {
    const float* x   = (const float*)d_in[0];
    const int*   ei  = (const int*)d_in[1];
    const float* W1l = (const float*)d_in[2];
    const float* b1  = (const float*)d_in[3];
    const float* W1r = (const float*)d_in[4];
    const float* W2l = (const float*)d_in[5];
    const float* b2  = (const float*)d_in[6];
    const float* W2r = (const float*)d_in[7];
    const float* Wc  = (const float*)d_in[8];
    const float* bc  = (const float*)d_in[9];

    const int E = in_sizes[1] / 2;
    const int* src = ei;
    const int* dst = ei + E;

    // workspace layout: [agg N*128][deg N][h1 N*128]
    float* A   = (float*)d_ws;
    float* deg = A + (size_t)N_NODES * FDIM;
    float* H   = deg + (size_t)N_NODES;

    const int ZT = 256;
    // zero agg + deg (contiguous, N*129 floats, divisible by 4)
    long n4ad = (long)N_NODES * (FDIM + 1) / 4;
    k_zero_f4<<<(int)((n4ad + ZT - 1) / ZT), ZT, 0, stream>>>((float4*)A, n4ad);

    k_deg<<<(E + ZT - 1) / ZT, ZT, 0, stream>>>(dst, deg, E);

    long sthreads = (long)E * 32;
    int sblocks = (int)((sthreads + ZT - 1) / ZT);

    // ---- layer 1 ----
    k_scatter<<<sblocks, ZT, 0, stream>>>(x, src, dst, A, E);
    int gblocks = N_NODES / 16;
    k_sage<<<gblocks, 256, 0, stream>>>(A, deg, x, W1l, W1r, b1, H);

    // ---- layer 2 ----
    long n4a = (long)N_NODES * FDIM / 4;
    k_zero_f4<<<(int)((n4a + ZT - 1) / ZT), ZT, 0, stream>>>((float4*)A, n4a);
    k_scatter<<<sblocks, ZT, 0, stream>>>(H, src, dst, A, E);
    k_sage<<<gblocks, 256, 0, stream>>>(A, deg, H, W2l, W2r, b2, /*in-place safe*/ A);

    // ---- classifier + log_softmax ----
    long cthreads = (long)N_NODES * 32;
    k_cls<<<(int)((cthreads + ZT - 1) / ZT), ZT, 0, stream>>>(A, Wc, bc, (float*)d_out);
}